// Separator_59760174956805
// MI455X (gfx1250) — compile-verified
//
#include <hip/hip_runtime.h>
#include <math.h>

typedef float v2f __attribute__((ext_vector_type(2)));
typedef float v4f __attribute__((ext_vector_type(4)));
typedef float v8f __attribute__((ext_vector_type(8)));

#define NFRAMES 2560
#define NBINS   2049
#define NCH     2
#define NFC     (NBINS * NCH)          /* 4098 (f,c) columns   */
#define EPS     1.1920928955078125e-7f /* FLT_EPSILON          */

#define TCHUNK          64
#define NTCHUNKS        (NFRAMES / TCHUNK)          /* 40  */
#define FC_TILES        ((NFC + 15) / 16)           /* 257 */
#define WAVES_PER_BLOCK 8
#define P1_BLOCKS       ((FC_TILES + WAVES_PER_BLOCK - 1) / WAVES_PER_BLOCK) /* 33 */

__device__ __forceinline__ float fast_sqrt(float a) {
    return __builtin_amdgcn_sqrtf(a);   // v_sqrt_f32
}
__device__ __forceinline__ float fast_rcp(float a) {
    return __builtin_amdgcn_rcpf(a);    // v_rcp_f32
}

// ---------------------------------------------------------------------------
// Kernel 0: zero the gain/weight accumulation table in workspace.
// ws layout: interleaved pairs per fc: ws[2*fc+0]=gain, ws[2*fc+1]=sum(vt^2)
// ---------------------------------------------------------------------------
__global__ void zero_ws_kernel(float* __restrict__ ws, int n) {
    int i = blockIdx.x * blockDim.x + threadIdx.x;
    if (i < n) ws[i] = 0.0f;
}

// ---------------------------------------------------------------------------
// Kernel 1: frame-axis reduction through the matrix pipe.
// Each wave owns a tile of 16 (f,c) columns and one 64-frame chunk.
// Per 4 frames it builds the f32 WMMA A/B operands directly from streaming
// loads (magnitude of x, source-sum of v) and issues two
// v_wmma_f32_16x16x4_f32 ops:
//   accG += Vx * Vt^T   -> diag = sum_t vx*vt   (gain)
//   accW += Vt * Vt^T   -> diag = sum_t vt*vt   (weights)
// Loads stay regular-temporal so v/x remain in the 192 MB L2 for pass 2.
// ---------------------------------------------------------------------------
__global__ __launch_bounds__(256)
void reduce_wmma_kernel(const float* __restrict__ v,
                        const float* __restrict__ x,
                        float* __restrict__ ws) {
    const int lane = threadIdx.x & 31;
    const int wave = threadIdx.x >> 5;
    const int tile = blockIdx.x * WAVES_PER_BLOCK + wave;
    if (tile >= FC_TILES) return;          // uniform per wave

    const int t0   = blockIdx.y * TCHUNK;
    const int m    = lane & 15;            // (f,c) row within tile
    const int fc   = tile * 16 + m;
    const int fcc  = (fc < NFC) ? fc : (NFC - 1);   // clamped address
    const float ok = (fc < NFC) ? 1.0f : 0.0f;      // zero padded rows
    const int tsel = (lane < 16) ? 0 : 2;  // half-wave K offset (K=0,1 | 2,3)

    v8f accG = {};   // gain matrix accumulator
    v8f accW = {};   // weight matrix accumulator

    for (int tc = 0; tc < TCHUNK; tc += 4) {
        const int ta = t0 + tc + tsel;     // K = 0 (or 2) component
        const int tb = ta + 1;             // K = 1 (or 3) component

        const v2f xa = *(const v2f*)(x + ((size_t)ta * NFC + fcc) * 2);
        const v2f xb = *(const v2f*)(x + ((size_t)tb * NFC + fcc) * 2);
        const v4f va = *(const v4f*)(v + ((size_t)ta * NFC + fcc) * 4);
        const v4f vb = *(const v4f*)(v + ((size_t)tb * NFC + fcc) * 4);

        const float vxa = fmaxf(EPS, fast_sqrt(xa.x * xa.x + xa.y * xa.y)) * ok;
        const float vxb = fmaxf(EPS, fast_sqrt(xb.x * xb.x + xb.y * xb.y)) * ok;
        const float vta = (va.x + va.y + va.z + va.w) * ok;
        const float vtb = (vb.x + vb.y + vb.z + vb.w) * ok;

        v2f A; A.x = vxa; A.y = vxb;       // Vx operand (16x4 f32)
        v2f B; B.x = vta; B.y = vtb;       // Vt operand (4x16 f32, same lane map)

        accG = __builtin_amdgcn_wmma_f32_16x16x4_f32(
                   false, A, false, B, (short)0, accG, false, false);
        accW = __builtin_amdgcn_wmma_f32_16x16x4_f32(
                   false, B, false, B, (short)0, accW, false, false);
    }

    // Diagonal element (d,d) of the 16x16 f32 D matrix lives at:
    //   d < 8 : VGPR d,   lane d
    //   d >= 8: VGPR d-8, lane d+16   (lanes 24..31)
    const bool owner = (lane < 8) || (lane >= 24);
    const int  comp  = lane & 7;
    float g = accG[0];
    float w = accW[0];
#pragma unroll
    for (int r = 1; r < 8; ++r) {
        if (comp == r) { g = accG[r]; w = accW[r]; }
    }
    const int dfc = tile * 16 + ((lane < 8) ? lane : (lane - 16));
    if (owner && dfc < NFC) {
        atomicAdd(&ws[2 * dfc + 0], g);
        atomicAdd(&ws[2 * dfc + 1], w);
    }
}

// ---------------------------------------------------------------------------
// Kernel 2: elementwise finalize. One thread per (t,f) handles both channels:
// non-temporal (last-use) loads of v (2x b128) and x (b128), regular load of
// the hot 32 KB gain/weight table, non-temporal b128 stores of the 20-float
// output chunk (5x v4f) so the 420 MB output stream doesn't flush L2.
// Output layout per (t,f): [c][ri][j] with j fastest (5 sources).
// ---------------------------------------------------------------------------
__global__ __launch_bounds__(256)
void finalize_kernel(const float* __restrict__ v,
                     const float* __restrict__ x,
                     const float* __restrict__ ws,
                     float* __restrict__ out) {
    const size_t tf = (size_t)blockIdx.x * blockDim.x + threadIdx.x;
    if (tf >= (size_t)NFRAMES * NBINS) return;
    const int f = (int)(tf % NBINS);

    const v4f gw = *(const v4f*)(ws + (size_t)f * 4);   // g0,w0,g1,w1 (hot, L2)
    const float s0 = gw.x * fast_rcp(EPS + gw.y);
    const float s1 = gw.z * fast_rcp(EPS + gw.w);

    const v4f xv = __builtin_nontemporal_load((const v4f*)(x + tf * 4)); // re0 im0 re1 im1
    const v4f v0 = __builtin_nontemporal_load((const v4f*)(v + tf * 8));
    const v4f v1 = __builtin_nontemporal_load((const v4f*)(v + tf * 8) + 1);

    // channel 0
    const float vt0 = (v0.x + v0.y + v0.z + v0.w) * s0;
    const float vx0 = fmaxf(EPS, fast_sqrt(xv.x * xv.x + xv.y * xv.y));
    const float vr0 = fmaxf(0.0f, vx0 - vt0);
    const float i0  = fast_rcp(EPS + vt0 + vr0);
    const float a0  = s0 * i0;
    const float m00 = v0.x * a0, m01 = v0.y * a0;
    const float m02 = v0.z * a0, m03 = v0.w * a0;
    const float m04 = vr0 * i0;

    // channel 1
    const float vt1 = (v1.x + v1.y + v1.z + v1.w) * s1;
    const float vx1 = fmaxf(EPS, fast_sqrt(xv.z * xv.z + xv.w * xv.w));
    const float vr1 = fmaxf(0.0f, vx1 - vt1);
    const float i1  = fast_rcp(EPS + vt1 + vr1);
    const float a1  = s1 * i1;
    const float m10 = v1.x * a1, m11 = v1.y * a1;
    const float m12 = v1.z * a1, m13 = v1.w * a1;
    const float m14 = vr1 * i1;

    const float re0 = xv.x, im0 = xv.y, re1 = xv.z, im1 = xv.w;

    v4f* o = (v4f*)(out + tf * 20);
    v4f o0, o1, o2, o3, o4;
    o0.x = re0 * m00; o0.y = re0 * m01; o0.z = re0 * m02; o0.w = re0 * m03;
    o1.x = re0 * m04; o1.y = im0 * m00; o1.z = im0 * m01; o1.w = im0 * m02;
    o2.x = im0 * m03; o2.y = im0 * m04; o2.z = re1 * m10; o2.w = re1 * m11;
    o3.x = re1 * m12; o3.y = re1 * m13; o3.z = re1 * m14; o3.w = im1 * m10;
    o4.x = im1 * m11; o4.y = im1 * m12; o4.z = im1 * m13; o4.w = im1 * m14;
    __builtin_nontemporal_store(o0, o + 0);
    __builtin_nontemporal_store(o1, o + 1);
    __builtin_nontemporal_store(o2, o + 2);
    __builtin_nontemporal_store(o3, o + 3);
    __builtin_nontemporal_store(o4, o + 4);
}

// ---------------------------------------------------------------------------
extern "C" void kernel_launch(void* const* d_in, const int* in_sizes, int n_in,
                              void* d_out, int out_size, void* d_ws, size_t ws_size,
                              hipStream_t stream) {
    const float* v = (const float*)d_in[0];
    const float* x = (const float*)d_in[1];
    float* ws  = (float*)d_ws;
    float* out = (float*)d_out;

    const int nacc = NFC * 2;  // 8196 floats
    zero_ws_kernel<<<(nacc + 255) / 256, 256, 0, stream>>>(ws, nacc);

    dim3 g1(P1_BLOCKS, NTCHUNKS);
    reduce_wmma_kernel<<<g1, 256, 0, stream>>>(v, x, ws);

    const size_t ntf = (size_t)NFRAMES * NBINS;
    finalize_kernel<<<(unsigned)((ntf + 255) / 256), 256, 0, stream>>>(v, x, ws, out);
}